// LinearBlendSkinning_44994077393500
// MI455X (gfx1250) — compile-verified
//
#include <hip/hip_runtime.h>
#include <hip/hip_bf16.h>

typedef __attribute__((ext_vector_type(2))) float v2f;
typedef __attribute__((ext_vector_type(8))) float v8f;

#define B_   64
#define P_   256
#define J_   128
#define K_   8
#define NCOL (7 * J_)   // 896

// ---------------------------------------------------------------------------
// Kernel 1: param = pose @ transform.T + transform_offsets  via WMMA f32 16x16x4
//   pose:      (64, 256) row-major
//   transform: (896, 256) row-major  -> B[k,n] = transform[n,k] (contiguous in k)
//   param:     (64, 896) row-major
// One wave = one 16x16 tile of param; 64 x V_WMMA_F32_16X16X4_F32 along K.
// ---------------------------------------------------------------------------
__global__ __launch_bounds__(32) void param_gemm_wmma(
    const float* __restrict__ pose,
    const float* __restrict__ transform,
    const float* __restrict__ offs,     // (896,)
    float* __restrict__ param)
{
    const int tileN = blockIdx.x;        // 0..55
    const int tileM = blockIdx.y;        // 0..3
    const int lane  = threadIdx.x;       // 0..31
    const int half  = lane >> 4;         // 0 or 1
    const int l15   = lane & 15;

    const int rowA = tileM * 16 + l15;   // A-matrix row held by this lane
    const int colB = tileN * 16 + l15;   // B-matrix column held by this lane
    const int kb   = half * 2;           // K-pair base within 4-wide K step

    const float* aptr = pose      + rowA * P_ + kb;
    const float* bptr = transform + colB * P_ + kb;

    v8f acc = {};
#pragma unroll 8
    for (int k0 = 0; k0 < P_; k0 += 4) {
        v2f a = *(const v2f*)(aptr + k0);
        v2f b = *(const v2f*)(bptr + k0);
        acc = __builtin_amdgcn_wmma_f32_16x16x4_f32(
            /*neg_a=*/false, a, /*neg_b=*/false, b,
            /*c_mod=*/(short)0, acc, /*reuse_a=*/false, /*reuse_b=*/false);
    }

    const float off  = offs[colB];
    const int mbase  = tileM * 16 + half * 8;   // D rows: mbase .. mbase+7
#pragma unroll
    for (int r = 0; r < 8; ++r)
        param[(mbase + r) * NCOL + colB] = acc[r] + off;
}

// ---------------------------------------------------------------------------
// Kernel 2: sequential skeleton solve + states_to_matrix, one thread per batch.
// Parent chain carried in registers (p==j-1 fast path); mats -> workspace.
// ---------------------------------------------------------------------------
__device__ __forceinline__ void qmul(const float a[4], const float b[4], float o[4]) {
    o[0] = a[3]*b[0] + a[0]*b[3] + a[1]*b[2] - a[2]*b[1];
    o[1] = a[3]*b[1] - a[0]*b[2] + a[1]*b[3] + a[2]*b[0];
    o[2] = a[3]*b[2] + a[0]*b[1] - a[1]*b[0] + a[2]*b[3];
    o[3] = a[3]*b[3] - a[0]*b[0] - a[1]*b[1] - a[2]*b[2];
}
__device__ __forceinline__ void qrot(const float q[4], const float v[3], float o[3]) {
    float tx = q[1]*v[2] - q[2]*v[1] + q[3]*v[0];
    float ty = q[2]*v[0] - q[0]*v[2] + q[3]*v[1];
    float tz = q[0]*v[1] - q[1]*v[0] + q[3]*v[2];
    o[0] = v[0] + 2.0f*(q[1]*tz - q[2]*ty);
    o[1] = v[1] + 2.0f*(q[2]*tx - q[0]*tz);
    o[2] = v[2] + 2.0f*(q[0]*ty - q[1]*tx);
}

__global__ __launch_bounds__(64) void skeleton_kernel(
    const float* __restrict__ param,          // (B, 896)
    const float* __restrict__ joint_offset,   // (J, 3)
    const float* __restrict__ joint_rotation, // (J, 4)
    const int*   __restrict__ joint_parents,  // (J,)
    const float* __restrict__ bind,           // (J, 8)
    float* __restrict__ states,               // ws: (B, J, 8)
    float* __restrict__ mat)                  // ws: (B, J, 12)
{
    const int b = blockIdx.x * blockDim.x + threadIdx.x;
    if (b >= B_) return;

    float pt[3] = {0.f, 0.f, 0.f};
    float pr[4] = {0.f, 0.f, 0.f, 1.f};
    float psc   = 1.f;

    for (int j = 0; j < J_; ++j) {
        const float* jp = param + b * NCOL + j * 7;

        // local transform
        float lt[3] = { jp[0] + joint_offset[j*3+0],
                        jp[1] + joint_offset[j*3+1],
                        jp[2] + joint_offset[j*3+2] };
        float s0,c0s,s1,c1s,s2,c2s;
        sincosf(0.5f * jp[3], &s0, &c0s);
        sincosf(0.5f * jp[4], &s1, &c1s);
        sincosf(0.5f * jp[5], &s2, &c2s);
        float qe[4] = { s0*c1s*c2s + c0s*s1*s2,
                        c0s*s1*c2s - s0*c1s*s2,
                        c0s*c1s*s2 + s0*s1*c2s,
                        c0s*c1s*c2s - s0*s1*s2 };
        const float jr[4] = { joint_rotation[j*4+0], joint_rotation[j*4+1],
                              joint_rotation[j*4+2], joint_rotation[j*4+3] };
        float lr[4]; qmul(jr, qe, lr);
        float ls = exp2f(jp[6]);

        // parent state
        const int p = joint_parents[j];
        float Pt[3], Pr[4], Ps;
        if (p < 0) {
            Pt[0]=Pt[1]=Pt[2]=0.f; Pr[0]=Pr[1]=Pr[2]=0.f; Pr[3]=1.f; Ps=1.f;
        } else if (p == j - 1) {
            Pt[0]=pt[0]; Pt[1]=pt[1]; Pt[2]=pt[2];
            Pr[0]=pr[0]; Pr[1]=pr[1]; Pr[2]=pr[2]; Pr[3]=pr[3]; Ps=psc;
        } else { // general tree fallback (not taken for a chain)
            const float* ps = states + ((size_t)b * J_ + p) * 8;
            Pt[0]=ps[0]; Pt[1]=ps[1]; Pt[2]=ps[2];
            Pr[0]=ps[3]; Pr[1]=ps[4]; Pr[2]=ps[5]; Pr[3]=ps[6]; Ps=ps[7];
        }

        // compose
        float gr[4]; qmul(Pr, lr, gr);
        float sl[3] = { lt[0]*Ps, lt[1]*Ps, lt[2]*Ps };
        float gtr[3]; qrot(Pr, sl, gtr);
        float gt[3] = { gtr[0]+Pt[0], gtr[1]+Pt[1], gtr[2]+Pt[2] };
        float gs = Ps * ls;

        float* st = states + ((size_t)b * J_ + j) * 8;
        st[0]=gt[0]; st[1]=gt[1]; st[2]=gt[2];
        st[3]=gr[0]; st[4]=gr[1]; st[5]=gr[2]; st[6]=gr[3]; st[7]=gs;

        // --- states_to_matrix fused ---
        const float* bs8 = bind + j * 8;
        float bq[4] = { bs8[3], bs8[4], bs8[5], bs8[6] };
        float inv = 1.0f / (bq[0]*bq[0] + bq[1]*bq[1] + bq[2]*bq[2] + bq[3]*bq[3]);
        float br[4] = { -bq[0]*inv, -bq[1]*inv, -bq[2]*inv, bq[3]*inv };
        float bsc = 1.0f / bs8[7];
        float nbt[3] = { -bs8[0], -bs8[1], -bs8[2] };
        float btv[3]; qrot(br, nbt, btv);
        btv[0]*=bsc; btv[1]*=bsc; btv[2]*=bsc;

        float tr[4]; qmul(gr, br, tr);
        float ts = gs * bsc;
        float sb[3] = { btv[0]*gs, btv[1]*gs, btv[2]*gs };
        float ttv[3]; qrot(gr, sb, ttv);
        float tt[3] = { ttv[0]+gt[0], ttv[1]+gt[1], ttv[2]+gt[2] };

        float x=tr[0], y=tr[1], z=tr[2], w=tr[3];
        float twx=2.f*x*w, twy=2.f*y*w, twz=2.f*z*w;
        float txx=2.f*x*x, txy=2.f*y*x, txz=2.f*z*x;
        float tyy=2.f*y*y, tyz=2.f*z*y, tzz=2.f*z*z;

        float* m = mat + ((size_t)b * J_ + j) * 12;
        m[0]  = (1.f-(tyy+tzz))*ts; m[1]  = (txy-twz)*ts;      m[2]  = (txz+twy)*ts;      m[3]  = tt[0];
        m[4]  = (txy+twz)*ts;       m[5]  = (1.f-(txx+tzz))*ts; m[6]  = (tyz-twx)*ts;      m[7]  = tt[1];
        m[8]  = (txz-twy)*ts;       m[9]  = (tyz+twx)*ts;       m[10] = (1.f-(txx+tyy))*ts; m[11] = tt[2];

        // carry state forward
        pt[0]=gt[0]; pt[1]=gt[1]; pt[2]=gt[2];
        pr[0]=gr[0]; pr[1]=gr[1]; pr[2]=gr[2]; pr[3]=gr[3]; psc=gs;
    }
}

// ---------------------------------------------------------------------------
// Kernel 3: linear blend skinning. grid = (ceil(V/256), B). Per-batch mats
// staged in LDS (6 KB), gathered with ds_load_b128; vectorized weight/index
// loads; non-temporal coalesced output stores (77 MB output > L2).
// ---------------------------------------------------------------------------
__global__ __launch_bounds__(256) void skin_kernel(
    const float* __restrict__ mat,      // (B, J, 12)
    const float* __restrict__ verts,    // (V, 3)
    const float* __restrict__ weights,  // (V, 8)
    const int*   __restrict__ indices,  // (V, 8)
    float* __restrict__ out,            // (B, V, 3)
    int V)
{
    __shared__ float smat[J_ * 12];     // 6144 B
    const int b = blockIdx.y;
    const float* msrc = mat + (size_t)b * J_ * 12;
    for (int i = threadIdx.x; i < J_ * 12; i += 256) smat[i] = msrc[i];
    __syncthreads();

    const int v = blockIdx.x * 256 + threadIdx.x;
    if (v >= V) return;

    const float vx = verts[v*3+0], vy = verts[v*3+1], vz = verts[v*3+2];
    const float4* wp = (const float4*)(weights + (size_t)v * K_);
    const int4*   ip = (const int4*)(indices + (size_t)v * K_);
    float4 w0 = wp[0], w1 = wp[1];
    int4   i0 = ip[0], i1 = ip[1];

    const int   idx[K_] = { i0.x, i0.y, i0.z, i0.w, i1.x, i1.y, i1.z, i1.w };
    const float wk [K_] = { w0.x, w0.y, w0.z, w0.w, w1.x, w1.y, w1.z, w1.w };

    float ox = 0.f, oy = 0.f, oz = 0.f;
#pragma unroll
    for (int k = 0; k < K_; ++k) {
        const float4* m = (const float4*)(smat + idx[k] * 12);
        float4 r0 = m[0], r1 = m[1], r2 = m[2];
        float w = wk[k];
        ox = fmaf(w, fmaf(r0.x, vx, fmaf(r0.y, vy, fmaf(r0.z, vz, r0.w))), ox);
        oy = fmaf(w, fmaf(r1.x, vx, fmaf(r1.y, vy, fmaf(r1.z, vz, r1.w))), oy);
        oz = fmaf(w, fmaf(r2.x, vx, fmaf(r2.y, vy, fmaf(r2.z, vz, r2.w))), oz);
    }

    float* op = out + ((size_t)b * V + v) * 3;
    __builtin_nontemporal_store(ox, op + 0);
    __builtin_nontemporal_store(oy, op + 1);
    __builtin_nontemporal_store(oz, op + 2);
}

// ---------------------------------------------------------------------------
extern "C" void kernel_launch(void* const* d_in, const int* in_sizes, int n_in,
                              void* d_out, int out_size, void* d_ws, size_t ws_size,
                              hipStream_t stream) {
    const float* pose            = (const float*)d_in[0];  // (64, 256)
    const float* transform       = (const float*)d_in[1];  // (896, 256)
    const float* transform_offs  = (const float*)d_in[2];  // (1, 896)
    const float* joint_offset    = (const float*)d_in[3];  // (128, 3)
    const float* joint_rotation  = (const float*)d_in[4];  // (128, 4)
    const float* bind_state      = (const float*)d_in[5];  // (1, 128, 8)
    const float* template_verts  = (const float*)d_in[6];  // (V, 3)
    const float* skin_weights    = (const float*)d_in[7];  // (V, 8)
    const int*   joint_parents   = (const int*)d_in[8];    // (128,)
    const int*   skin_indices    = (const int*)d_in[9];    // (V, 8)
    float* out = (float*)d_out;

    const int V = in_sizes[6] / 3;

    // workspace layout (floats)
    float* ws     = (float*)d_ws;
    float* param  = ws;                                   // 64*896
    float* states = param  + (size_t)B_ * NCOL;           // 64*128*8
    float* mats   = states + (size_t)B_ * J_ * 8;         // 64*128*12

    // 1) param GEMM via WMMA fp32
    param_gemm_wmma<<<dim3(NCOL / 16, B_ / 16), 32, 0, stream>>>(
        pose, transform, transform_offs, param);

    // 2) sequential skeleton solve + matrix build (one thread per batch)
    skeleton_kernel<<<1, 64, 0, stream>>>(
        param, joint_offset, joint_rotation, joint_parents, bind_state,
        states, mats);

    // 3) skinning: bandwidth-bound main phase
    skin_kernel<<<dim3((V + 255) / 256, B_), 256, 0, stream>>>(
        mats, template_verts, skin_weights, skin_indices, out, V);
}